// SymmetricAttentionSplit_60954175864854
// MI455X (gfx1250) — compile-verified
//
#include <hip/hip_runtime.h>
#include <hip/hip_bf16.h>

// Problem constants (match reference)
constexpr int B_   = 16;
constexpr int N_   = 128;
constexpr int NC_  = 32;
constexpr int H_   = 256;
constexpr int A_   = 64;
constexpr int DEG_ = 3;
constexpr int MTOT = B_ * N_;           // 2048 flattened rows of x
// SCALE = 4096 = 16^3 -> fold 1/16 into each y[d]
constexpr float YPRE = 1.0f / 16.0f;

typedef float v2f __attribute__((ext_vector_type(2)));
typedef float v4f __attribute__((ext_vector_type(4)));
typedef float v8f __attribute__((ext_vector_type(8)));

__device__ __forceinline__ v8f wmma_f32_4(v2f a, v2f b, v8f c) {
  // D = A(16x4 f32) x B(4x16 f32) + C(16x16 f32)
  return __builtin_amdgcn_wmma_f32_16x16x4_f32(
      /*neg_a=*/false, a, /*neg_b=*/false, b,
      /*c_mod=*/(short)0, c, /*reuse_a=*/false, /*reuse_b=*/false);
}

// tanh-approx GELU, branch-free: tanh(u) = 1 - 2/(1+exp(2u))
// (exp->inf gives tanh->1, exp->0 gives tanh->-1; no NaN at extremes)
__device__ __forceinline__ float gelu_tanh(float v) {
  const float c0 = 0.7978845608028654f;  // sqrt(2/pi)
  const float c1 = 0.044715f;
  float u = c0 * (v + c1 * v * v * v);
  float e = __expf(2.0f * u);
  float t = 1.0f - 2.0f / (e + 1.0f);
  return 0.5f * v * (1.0f + t);
}

// ---------------------------------------------------------------------------
// Kernel 1: masked mean-pool of condition + dv[d] = pooled @ Wc[d]
// grid (B_, DEG_), block H_ threads. Also writes dvs output tail.
// ---------------------------------------------------------------------------
__global__ __launch_bounds__(H_)
void pool_dv_kernel(const float* __restrict__ cond,
                    const float* __restrict__ cmask,
                    const float* __restrict__ Wc,
                    float* __restrict__ dv_ws,
                    float* __restrict__ dvs_out) {
  int b = blockIdx.x, d = blockIdx.y, t = threadIdx.x;
  __shared__ float pooled[H_];
  float s = 0.f, ms = 0.f;
  for (int j = 0; j < NC_; ++j) {
    float mk = cmask[b * NC_ + j];
    s  += cond[(b * NC_ + j) * H_ + t] * mk;
    ms += mk;
  }
  pooled[t] = s / fmaxf(ms, 1.0f);
  __syncthreads();
  const float* W = Wc + (size_t)d * H_ * H_;
  float acc = 0.f;
  for (int k = 0; k < H_; ++k) acc += pooled[k] * W[k * H_ + t];
  int idx = (d * B_ + b) * H_ + t;
  dv_ws[idx]   = acc;
  dvs_out[idx] = acc;   // second reference output [DEG,B,H]
}

// ---------------------------------------------------------------------------
// Kernel 2: y[d] = gelu(x @ W1[d] + b1[d] + dv[d]) @ Wlin[d] * mask * (1/16)
// grid (MTOT/16, DEG_), block 256 (8 waves). f32 WMMA 16x16x4.
// Single shared 16x260 tile: holds x, then (after barrier) h.
// ---------------------------------------------------------------------------
__global__ __launch_bounds__(256)
void encoder_kernel(const float* __restrict__ x,
                    const float* __restrict__ xmask,
                    const float* __restrict__ W1,
                    const float* __restrict__ b1,
                    const float* __restrict__ Wlin,
                    const float* __restrict__ blin,
                    const float* __restrict__ dv_ws,
                    float* __restrict__ y_ws) {
  constexpr int LSTR = 260;  // 256 + 4 pad: row-to-row bank offset of 4
  int tileM = blockIdx.x;            // 16-row tile of flattened M
  int d     = blockIdx.y;
  int tid   = threadIdx.x;
  int wave  = tid >> 5, lane = tid & 31;
  int r0    = tileM * 16;
  int bidx  = r0 / N_;               // whole tile shares one batch (128 % 16 == 0)

  __shared__ float ts[16][LSTR];     // x tile, then h tile

  // stage x tile as float4 (row stride 260 floats keeps 16B alignment)
  for (int i = tid; i < 16 * (H_ / 4); i += 256) {
    int m = i >> 6, k4 = (i & 63) << 2;
    v4f v = *(const v4f*)&x[(size_t)(r0 + m) * H_ + k4];
    *(v4f*)&ts[m][k4] = v;
  }
  __syncthreads();

  int mA = lane & 15;
  int kk = (lane >> 4) << 1;         // 0 for lanes 0-15, 2 for lanes 16-31
  int mBase = (lane >> 4) << 3;      // C/D row base per half-wave

  // ---- h = gelu(x @ W1 + b1 + dv): 16 column tiles, 2 per wave ----
  const float* W1d = W1 + (size_t)d * H_ * H_;
  v8f haccs[2];
  for (int rep = 0; rep < 2; ++rep) {
    int col = (wave + rep * 8) * 16 + (lane & 15);
    v8f acc = {0.f, 0.f, 0.f, 0.f, 0.f, 0.f, 0.f, 0.f};
    for (int k0 = 0; k0 < H_; k0 += 4) {
      v2f a, bb;
      a.x  = ts[mA][k0 + kk];
      a.y  = ts[mA][k0 + kk + 1];
      bb.x = W1d[(size_t)(k0 + kk) * H_ + col];
      bb.y = W1d[(size_t)(k0 + kk + 1) * H_ + col];
      acc = wmma_f32_4(a, bb, acc);
    }
    haccs[rep] = acc;
  }
  __syncthreads();                   // all x reads done; reuse ts for h
  for (int rep = 0; rep < 2; ++rep) {
    int col = (wave + rep * 8) * 16 + (lane & 15);
    float bias = b1[d * H_ + col] + dv_ws[(d * B_ + bidx) * H_ + col];
    for (int v = 0; v < 8; ++v)
      ts[v + mBase][col] = gelu_tanh(haccs[rep][v] + bias);
  }
  __syncthreads();

  // ---- y = h @ Wlin + blin, masked + prescaled: 4 col tiles, waves 0-3 ----
  if (wave < 4) {
    const float* Wl = Wlin + (size_t)d * H_ * A_;
    int col = wave * 16 + (lane & 15);
    v8f acc = {0.f, 0.f, 0.f, 0.f, 0.f, 0.f, 0.f, 0.f};
    for (int k0 = 0; k0 < H_; k0 += 4) {
      v2f a, bb;
      a.x  = ts[mA][k0 + kk];
      a.y  = ts[mA][k0 + kk + 1];
      bb.x = Wl[(size_t)(k0 + kk) * A_ + col];
      bb.y = Wl[(size_t)(k0 + kk + 1) * A_ + col];
      acc = wmma_f32_4(a, bb, acc);
    }
    float bias = blin[d * A_ + col];
    for (int v = 0; v < 8; ++v) {
      int row = r0 + v + mBase;
      y_ws[((size_t)d * MTOT + row) * A_ + col] =
          (acc[v] + bias) * xmask[row] * YPRE;
    }
  }
}

// ---------------------------------------------------------------------------
// Kernel 3: symmetrized degree-3 contraction, written directly (once).
// out[b,x,y,z] = c01(y)·Y2[z] + c02(y)·Y1[z] + c12(y)·Y0[z], where
//   cab[y,i] = ya[x,i]*yb[y,i] + yb[x,i]*ya[y,i]   (sum over i via WMMA)
// grid (N_, N_/16, B_), block 256 (8 waves, one 16-wide z tile each).
// ---------------------------------------------------------------------------
__global__ __launch_bounds__(256)
void contract_kernel(const float* __restrict__ y_ws,
                     float* __restrict__ out) {
  constexpr int CSTR = 68;   // 64 + 4 pad (272B row stride, 16B aligned)
  int xi = blockIdx.x;       // x row
  int yt = blockIdx.y;       // 16-row y tile
  int b  = blockIdx.z;
  int tid = threadIdx.x;
  int wave = tid >> 5, lane = tid & 31;

  __shared__ float xr[3][64];        // y_d[x, :]
  __shared__ float cm[3][16][CSTR];  // the three A-matrices (16 x 64)

  for (int i = tid; i < 3 * 64; i += 256) {
    int d = i >> 6, k = i & 63;
    xr[d][k] = y_ws[((size_t)d * MTOT + b * N_ + xi) * A_ + k];
  }
  __syncthreads();

  int ybase = yt * 16;
  // build c-matrices, 4 consecutive k per thread -> b128 loads/stores
  for (int e = tid; e < 3 * 16 * (A_ / 4); e += 256) {
    int pos = e << 2;               // element index, k-multiple-of-4
    int t   = pos >> 10;
    int rem = pos & 1023;
    int yy  = rem >> 6, k = rem & 63;
    int yrow = ybase + yy;
    v4f y0v = *(const v4f*)&y_ws[((size_t)0 * MTOT + b * N_ + yrow) * A_ + k];
    v4f y1v = *(const v4f*)&y_ws[((size_t)1 * MTOT + b * N_ + yrow) * A_ + k];
    v4f y2v = *(const v4f*)&y_ws[((size_t)2 * MTOT + b * N_ + yrow) * A_ + k];
    v4f res;
    for (int j = 0; j < 4; ++j) {
      float a0 = xr[0][k + j], a1 = xr[1][k + j], a2 = xr[2][k + j];
      if      (t == 0) res[j] = a0 * y1v[j] + a1 * y0v[j];  // pairs with Y2
      else if (t == 1) res[j] = a0 * y2v[j] + a2 * y0v[j];  // pairs with Y1
      else             res[j] = a1 * y2v[j] + a2 * y1v[j];  // pairs with Y0
    }
    *(v4f*)&cm[t][yy][k] = res;
  }
  __syncthreads();

  int zbase = wave * 16;
  int mA = lane & 15;
  int kk = (lane >> 4) << 1;
  int zcol = zbase + (lane & 15);
  v8f acc = {0.f, 0.f, 0.f, 0.f, 0.f, 0.f, 0.f, 0.f};

  for (int t = 0; t < 3; ++t) {
    const float* yB = y_ws + ((size_t)(2 - t) * MTOT + b * N_) * A_;
    for (int k0 = 0; k0 < A_; k0 += 4) {
      v2f a, bb;
      a.x = cm[t][mA][k0 + kk];
      a.y = cm[t][mA][k0 + kk + 1];
      bb  = *(const v2f*)&yB[(size_t)zcol * A_ + k0 + kk];  // 8B-aligned pair
      acc = wmma_f32_4(a, bb, acc);
    }
  }

  size_t outbase = (((size_t)b * N_ + xi) * N_) * N_;
  int mBase = (lane >> 4) << 3;
  for (int v = 0; v < 8; ++v) {
    out[outbase + (size_t)(ybase + v + mBase) * N_ + zbase + (lane & 15)] = acc[v];
  }
}

// ---------------------------------------------------------------------------
extern "C" void kernel_launch(void* const* d_in, const int* in_sizes, int n_in,
                              void* d_out, int out_size, void* d_ws, size_t ws_size,
                              hipStream_t stream) {
  const float* x     = (const float*)d_in[0];
  const float* xmask = (const float*)d_in[1];
  const float* cond  = (const float*)d_in[2];
  const float* cmask = (const float*)d_in[3];
  const float* W1    = (const float*)d_in[4];
  const float* b1    = (const float*)d_in[5];
  const float* Wc    = (const float*)d_in[6];
  const float* Wlin  = (const float*)d_in[7];
  const float* blin  = (const float*)d_in[8];

  float* out     = (float*)d_out;
  float* dvs_out = out + (size_t)B_ * N_ * N_ * N_;   // tuple tail [DEG,B,H]

  float* dv_ws = (float*)d_ws;                        // DEG*B*H floats
  float* y_ws  = dv_ws + (size_t)DEG_ * B_ * H_;      // DEG*MTOT*A floats (~1.5MB)

  pool_dv_kernel<<<dim3(B_, DEG_), H_, 0, stream>>>(cond, cmask, Wc, dv_ws, dvs_out);
  encoder_kernel<<<dim3(MTOT / 16, DEG_), 256, 0, stream>>>(
      x, xmask, W1, b1, Wlin, blin, dv_ws, y_ws);
  contract_kernel<<<dim3(N_, N_ / 16, B_), 256, 0, stream>>>(y_ws, out);
}